// MambaLikeLinearAttention_3985729651466
// MI455X (gfx1250) — compile-verified
//
#include <hip/hip_runtime.h>
#include <hip/hip_bf16.h>

typedef __attribute__((ext_vector_type(2))) float v2f;
typedef __attribute__((ext_vector_type(8))) float v8f;

#define BB 16
#define NN 4096
#define CC 384
#define NHEAD 12
#define DHEAD 32
#define KMAX 96   // C/4
#define LN10000 9.210340371976184f

__device__ __forceinline__ v8f wmma4(v2f a, v2f b, v8f c) {
    // D = A(16x4) * B(4x16) + C(16x16), fp32
    return __builtin_amdgcn_wmma_f32_16x16x4_f32(
        /*neg_a=*/false, a, /*neg_b=*/false, b,
        /*c_mod=*/(short)0, c, /*reuse_a=*/false, /*reuse_b=*/false);
}

// RoPE angle for global pair index p (0..191) and token n (0..4095).
__device__ __forceinline__ void rope_cs(int p, int n, float& c, float& s) {
    int k = (p < KMAX) ? p : (p - KMAX);
    float coord = (p < KMAX) ? (float)(n >> 6) : (float)(n & 63);
    float theta = __expf(-(float)k * (LN10000 / (float)KMAX));
    float ang = coord * theta;
    __sincosf(ang, &s, &c);
}

__device__ __forceinline__ float elu1(float v) {
    return v >= 0.0f ? (v + 1.0f) : __expf(v);
}

// ---------------------------------------------------------------------------
// Kernel A: qk = x @ qk_w.T + b ; q_elu = elu(qk[:, :C])+1 ; k_elu = elu(qk[:, C:])+1
// also accumulates k_sum[b][c] = sum_n k_elu  (for z)
// Block: 256 thr (8 waves). Block tile: 64 rows x 256 cols.
// Wave: 2 col-tiles x 4 row-tiles (8 WMMA accumulators) -> 0.75 loads/WMMA.
// Explicit 2-stage software pipeline (branch-free via chunk peeling): next-step
// fragments are loaded before the current step's WMMAs consume their inputs.
// ---------------------------------------------------------------------------

// One pipelined K-step: load fragments for K=kn into n*, run 8 WMMAs on c*.
#define QK_LOAD_NEXT(kn)                         \
    nb0 = *(const v2f*)(pw0 + (kn));             \
    nb1 = *(const v2f*)(pw1 + (kn));             \
    na0 = *(const v2f*)(px0 + (kn));             \
    na1 = *(const v2f*)(px1 + (kn));             \
    na2 = *(const v2f*)(px2 + (kn));             \
    na3 = *(const v2f*)(px3 + (kn));

#define QK_WMMA8()                               \
    acc[0][0] = wmma4(ca0, cb0, acc[0][0]);      \
    acc[0][1] = wmma4(ca0, cb1, acc[0][1]);      \
    acc[1][0] = wmma4(ca1, cb0, acc[1][0]);      \
    acc[1][1] = wmma4(ca1, cb1, acc[1][1]);      \
    acc[2][0] = wmma4(ca2, cb0, acc[2][0]);      \
    acc[2][1] = wmma4(ca2, cb1, acc[2][1]);      \
    acc[3][0] = wmma4(ca3, cb0, acc[3][0]);      \
    acc[3][1] = wmma4(ca3, cb1, acc[3][1]);

#define QK_ROTATE()                              \
    cb0 = nb0; cb1 = nb1;                        \
    ca0 = na0; ca1 = na1; ca2 = na2; ca3 = na3;

__global__ void __launch_bounds__(256) qk_kernel(
                          const float* __restrict__ x,
                          const float* __restrict__ qk_w,
                          const float* __restrict__ qk_b,
                          float* __restrict__ q_elu,
                          float* __restrict__ k_elu,
                          float* __restrict__ k_sum) {
    const int tid  = threadIdx.x;
    const int lane = tid & 31;
    const int wave = tid >> 5;
    const int hi   = lane >> 4;   // 0: K{0,1} rows M0-15 ; 1: K{2,3}
    const int lo   = lane & 15;

    const long m_base = (long)blockIdx.x * 64;           // global row (b*N+n)
    const int  j0     = blockIdx.y * 256 + wave * 32;    // first of 2 col-tiles
    const int  jc0    = j0 + lo;
    const int  jc1    = j0 + 16 + lo;

    // Per-lane base pointers (include the hi*2 K-offset of the fragment layout)
    const float* pw0 = qk_w + (long)jc0 * CC + hi * 2;
    const float* pw1 = qk_w + (long)jc1 * CC + hi * 2;
    const float* px0 = x + (m_base + 0 * 16 + lo) * CC + hi * 2;
    const float* px1 = x + (m_base + 1 * 16 + lo) * CC + hi * 2;
    const float* px2 = x + (m_base + 2 * 16 + lo) * CC + hi * 2;
    const float* px3 = x + (m_base + 3 * 16 + lo) * CC + hi * 2;

    v8f acc[4][2] = {};

    // Pipeline stage registers: current + next fragments
    v2f cb0 = *(const v2f*)pw0;
    v2f cb1 = *(const v2f*)pw1;
    v2f ca0 = *(const v2f*)px0;
    v2f ca1 = *(const v2f*)px1;
    v2f ca2 = *(const v2f*)px2;
    v2f ca3 = *(const v2f*)px3;
    v2f nb0, nb1, na0, na1, na2, na3;

    // Chunks 0..4: every step has a valid next step (branch-free bodies).
    #pragma unroll 1
    for (int kc = 0; kc < CC - 64; kc += 64) {
        // Prefetch next chunk: hi half-split covers both 128B lines per row.
        const int pf = kc + 64 + hi * 32;
        __builtin_prefetch(px0 + pf, 0, 3);
        __builtin_prefetch(px1 + pf, 0, 3);
        __builtin_prefetch(px2 + pf, 0, 3);
        __builtin_prefetch(px3 + pf, 0, 3);
        __builtin_prefetch(pw0 + pf, 0, 3);
        __builtin_prefetch(pw1 + pf, 0, 3);
        #pragma unroll
        for (int ks = 0; ks < 64; ks += 4) {
            const int kn = kc + ks + 4;
            QK_LOAD_NEXT(kn)
            QK_WMMA8()
            QK_ROTATE()
        }
    }
    // Final chunk (kc = CC-64): steps 0..14 pipelined, last step compute-only.
    {
        #pragma unroll
        for (int ks = 0; ks < 60; ks += 4) {
            const int kn = (CC - 64) + ks + 4;
            QK_LOAD_NEXT(kn)
            QK_WMMA8()
            QK_ROTATE()
        }
        QK_WMMA8()   // K = CC-4
    }

    const int bidx = (int)(m_base >> 12);       // / 4096 (64 | 4096 -> single b)

    #pragma unroll
    for (int ct = 0; ct < 2; ++ct) {
        const int jc   = j0 + ct * 16 + lo;
        const float bj = qk_b[jc];
        const bool is_q = (jc < CC);            // uniform per (wave, ct): 32 | 384
        float* dst      = is_q ? q_elu : k_elu;
        const int jcol  = is_q ? jc : (jc - CC);

        #pragma unroll
        for (int rt = 0; rt < 4; ++rt) {
            float part = 0.0f;
            #pragma unroll
            for (int e = 0; e < 8; ++e) {
                float val = elu1(acc[rt][ct][e] + bj);
                long row = m_base + rt * 16 + e + hi * 8;   // C/D: M = e + hi*8
                dst[row * CC + jcol] = val;
                part += val;
            }
            if (!is_q) atomicAdd(&k_sum[bidx * CC + jcol], part);
        }
    }
}

// ---------------------------------------------------------------------------
// Kernel B: kv[b,h,d,e] = (1/N) * sum_n rope(k_elu)[n,d] * x[n,e]   (per head)
// Split-K over tokens (grid.y = 16, 256 tokens each), f32 atomic reduction.
// One wave per block; 32x32 output = 2x2 WMMA tiles.
// ---------------------------------------------------------------------------
__global__ void kv_kernel(const float* __restrict__ x,
                          const float* __restrict__ k_elu,
                          float* __restrict__ kv) {
    const int lane = threadIdx.x & 31;
    const int hi   = lane >> 4;
    const int lo   = lane & 15;
    const int bh = blockIdx.x;
    const int b  = bh / NHEAD, h = bh % NHEAD;
    const int cb = h * DHEAD;
    const long xbase = (long)b * NN * CC;
    const int t0 = blockIdx.y * 256;

    v8f acc[2][2] = {};

    for (int n0 = t0; n0 < t0 + 256; n0 += 4) {
        // A fragments: A[d][n] = k_rope[n][d]; lane = channel (M), VGPRs = tokens (K)
        v2f a[2];
        #pragma unroll
        for (int di = 0; di < 2; ++di) {
            const int ch = di * 16 + lo;               // local channel 0..31
            const int p  = h * 16 + (ch >> 1);         // global rope pair
            const int tokA = n0 + hi * 2;
            float va[2];
            #pragma unroll
            for (int t = 0; t < 2; ++t) {
                const int tok = tokA + t;
                v2f pr = *(const v2f*)(k_elu + xbase + (long)tok * CC + cb + (ch & ~1));
                float c, s; rope_cs(p, tok, c, s);
                va[t] = (ch & 1) ? (s * pr.x + c * pr.y) : (c * pr.x - s * pr.y);
            }
            a[di].x = va[0]; a[di].y = va[1];
        }
        // B fragments: B[n][e] = v = x
        v2f bv[2];
        #pragma unroll
        for (int ei = 0; ei < 2; ++ei) {
            const int e = ei * 16 + lo;
            const int tokB = n0 + hi * 2;
            bv[ei].x = x[xbase + (long)tokB * CC + cb + e];
            bv[ei].y = x[xbase + (long)(tokB + 1) * CC + cb + e];
        }
        #pragma unroll
        for (int di = 0; di < 2; ++di)
            #pragma unroll
            for (int ei = 0; ei < 2; ++ei)
                acc[di][ei] = wmma4(a[di], bv[ei], acc[di][ei]);
    }

    const float scale = 1.0f / (float)NN;
    float* kvh = kv + (long)bh * (DHEAD * DHEAD);
    #pragma unroll
    for (int di = 0; di < 2; ++di)
        #pragma unroll
        for (int ei = 0; ei < 2; ++ei)
            #pragma unroll
            for (int e = 0; e < 8; ++e) {
                const int d   = di * 16 + e + hi * 8;
                const int col = ei * 16 + lo;
                atomicAdd(&kvh[d * DHEAD + col], acc[di][ei][e] * scale);
            }
}

// ---------------------------------------------------------------------------
// Kernel C: out[n, h, :] = z[n,h] * ( rope(q_elu)[n, :] @ kv[h] )
//           z = 1 / (q_elu . k_mean + 1e-6)
// Block: 256 thr (8 waves); wave = one 16-token M-tile, 2 e-tiles. kv in regs.
// ---------------------------------------------------------------------------
__global__ void out_kernel(const float* __restrict__ q_elu,
                           const float* __restrict__ k_sum,
                           const float* __restrict__ kv,
                           float* __restrict__ out) {
    const int tid  = threadIdx.x;
    const int lane = tid & 31;
    const int wave = tid >> 5;
    const int hi   = lane >> 4;
    const int lo   = lane & 15;
    const int bh = blockIdx.x;
    const int b  = bh / NHEAD, h = bh % NHEAD;
    const int cb = h * DHEAD;
    const int m0 = blockIdx.y * 128 + wave * 16;
    const long rowbase = (long)b * NN;

    // Preload kv B fragments: B[k][e] = kv[h][k][e]
    v2f kvb[8][2];
    const float* kvh = kv + (long)bh * (DHEAD * DHEAD);
    #pragma unroll
    for (int ks = 0; ks < 8; ++ks) {
        const int kk = ks * 4 + hi * 2;
        #pragma unroll
        for (int ei = 0; ei < 2; ++ei) {
            const int e = ei * 16 + lo;
            kvb[ks][ei].x = kvh[kk * DHEAD + e];
            kvb[ks][ei].y = kvh[(kk + 1) * DHEAD + e];
        }
    }

    // z for token m0+lo (lanes 16-31 duplicate lanes 0-15; shfl reads 0..15)
    const int tok = m0 + lo;
    float dot = 0.0f;
    {
        const float* qrow = q_elu + (rowbase + tok) * CC + cb;
        const float* km   = k_sum + b * CC + cb;
        #pragma unroll
        for (int c4 = 0; c4 < DHEAD; c4 += 4) {
            float4 qv  = *(const float4*)(qrow + c4);
            float4 km4 = *(const float4*)(km + c4);
            dot += qv.x * km4.x + qv.y * km4.y + qv.z * km4.z + qv.w * km4.w;
        }
    }
    const float zv = 1.0f / (dot * (1.0f / (float)NN) + 1e-6f);

    v8f acc[2] = {};
    #pragma unroll
    for (int ks = 0; ks < 8; ++ks) {
        const int kk = ks * 4 + hi * 2;               // even -> (kk,kk+1) is a rope pair
        const int p  = h * 16 + (kk >> 1);
        v2f q2 = *(const v2f*)(q_elu + (rowbase + tok) * CC + cb + kk);
        float c, s; rope_cs(p, tok, c, s);
        v2f a;
        a.x = c * q2.x - s * q2.y;                    // K=kk   (rotated re)
        a.y = s * q2.x + c * q2.y;                    // K=kk+1 (rotated im)
        acc[0] = wmma4(a, kvb[ks][0], acc[0]);
        acc[1] = wmma4(a, kvb[ks][1], acc[1]);
    }

    #pragma unroll
    for (int e = 0; e < 8; ++e) {
        const int M  = e + hi * 8;
        const float zm = __shfl(zv, M, 32);
        const long row = rowbase + m0 + M;
        out[row * CC + cb + 0  + lo] = acc[0][e] * zm;
        out[row * CC + cb + 16 + lo] = acc[1][e] * zm;
    }
}

// ---------------------------------------------------------------------------
// Kernel D: out += depthwise 3x3 conv(x) + lepe_b   (SAME padding)
// ---------------------------------------------------------------------------
__global__ void lepe_kernel(const float* __restrict__ x,
                            const float* __restrict__ w,
                            const float* __restrict__ bias,
                            float* __restrict__ out) {
    const long idx = (long)blockIdx.x * blockDim.x + threadIdx.x;
    const long total = (long)BB * NN * CC;
    if (idx >= total) return;
    const int c  = (int)(idx % CC);
    const long bn = idx / CC;
    const int n = (int)(bn & (NN - 1));
    const int b = (int)(bn >> 12);
    const int y = n >> 6, xx = n & 63;
    const float* xb = x + (long)b * NN * CC;
    float acc = bias[c];
    #pragma unroll
    for (int dy = -1; dy <= 1; ++dy) {
        const int yy = y + dy;
        if (yy < 0 || yy > 63) continue;
        #pragma unroll
        for (int dx = -1; dx <= 1; ++dx) {
            const int xw = xx + dx;
            if (xw < 0 || xw > 63) continue;
            acc += xb[(long)(yy * 64 + xw) * CC + c] * w[c * 9 + (dy + 1) * 3 + (dx + 1)];
        }
    }
    out[idx] += acc;
}

// ---------------------------------------------------------------------------
extern "C" void kernel_launch(void* const* d_in, const int* in_sizes, int n_in,
                              void* d_out, int out_size, void* d_ws, size_t ws_size,
                              hipStream_t stream) {
    const float* x      = (const float*)d_in[0];
    const float* qk_w   = (const float*)d_in[1];
    const float* qk_b   = (const float*)d_in[2];
    const float* lepe_w = (const float*)d_in[3];
    const float* lepe_b = (const float*)d_in[4];
    float* out = (float*)d_out;

    float* ws = (float*)d_ws;
    const long NELEM = (long)BB * NN * CC;          // 25,165,824
    float* q_elu = ws;
    float* k_elu = ws + NELEM;
    float* k_sum = ws + 2 * NELEM;                  // 16*384
    float* kv    = k_sum + (long)BB * CC;           // 16*12*32*32

    // zero k_sum + kv (contiguous)
    hipMemsetAsync(k_sum, 0,
                   ((long)BB * CC + (long)BB * NHEAD * DHEAD * DHEAD) * sizeof(float),
                   stream);

    // A: projection GEMM (M=65536, K=384, N=768) + elu+1 + k_sum
    qk_kernel<<<dim3(1024, 3), 256, 0, stream>>>(x, qk_w, qk_b, q_elu, k_elu, k_sum);

    // B: kv = k_rope^T v / N  (192 heads x splitK 16)
    kv_kernel<<<dim3(BB * NHEAD, 16), 32, 0, stream>>>(x, k_elu, kv);

    // C: out = z * (q_rope @ kv)
    out_kernel<<<dim3(BB * NHEAD, 32), 256, 0, stream>>>(q_elu, k_sum, kv, out);

    // D: out += LePE depthwise conv + bias
    const long total = NELEM;
    lepe_kernel<<<(unsigned)((total + 255) / 256), 256, 0, stream>>>(x, lepe_w, lepe_b, out);
}